// PointNet_cov_3513283248944
// MI455X (gfx1250) — compile-verified
//
#include <hip/hip_runtime.h>
#include <hip/hip_bf16.h>
#include <math.h>

typedef float v2f __attribute__((ext_vector_type(2)));
typedef float v8f __attribute__((ext_vector_type(8)));

#define BN_EPS 1e-5f
#define KNN_K 20
#define PN_B 4
#define PN_N 8192
#define PN_M (PN_B * PN_N)

// ---------- monotone float<->uint encoding for atomic min/max ----------
__device__ __forceinline__ unsigned enc_f32(float f) {
    unsigned u = __float_as_uint(f);
    return (u & 0x80000000u) ? ~u : (u | 0x80000000u);
}
__device__ __forceinline__ float dec_f32(unsigned u) {
    unsigned v = (u & 0x80000000u) ? (u & 0x7fffffffu) : ~u;
    return __uint_as_float(v);
}

// ---------- init: zero BN accumulators, set min/max sentinels ----------
__global__ void init_stats_kernel(float* stats, int nStats,
                                  unsigned* mmax, unsigned* mmin, int nMM) {
    int i = blockIdx.x * blockDim.x + threadIdx.x;
    if (i < nStats) stats[i] = 0.0f;
    if (i < nMM) {
        mmax[i] = enc_f32(-INFINITY);
        mmin[i] = enc_f32(INFINITY);
    }
}

// ---------- kNN + covariance feature builder ----------
// x: [B,3,N], mask: [B,N] (bool bytes), h0: [12, B*N] channel-major
__global__ __launch_bounds__(256)
void knn_cov_kernel(const float* __restrict__ x, const unsigned char* __restrict__ mask,
                    float* __restrict__ h0) {
    const int gid = blockIdx.x * blockDim.x + threadIdx.x;   // one point per thread
    const int b = gid / PN_N;                                // blocks never straddle batches
    const int n = gid - b * PN_N;
    const float* xb = x + (size_t)b * 3 * PN_N;
    const unsigned char* mb = mask + (size_t)b * PN_N;

    const float xn0 = xb[n], xn1 = xb[PN_N + n], xn2 = xb[2 * PN_N + n];

    float bestv[KNN_K];
    int   besti[KNN_K];
#pragma unroll
    for (int j = 0; j < KNN_K; ++j) { bestv[j] = -INFINITY; besti[j] = n; }

    __shared__ float sx0[256], sx1[256], sx2[256];
    __shared__ unsigned char sm[256];

    for (int m0 = 0; m0 < PN_N; m0 += 256) {
        __syncthreads();
        const int t = threadIdx.x;
        sx0[t] = xb[m0 + t];
        sx1[t] = xb[PN_N + m0 + t];
        sx2[t] = xb[2 * PN_N + m0 + t];
        sm[t]  = mb[m0 + t];
        __syncthreads();
        for (int tt = 0; tt < 256; ++tt) {
            const float dx = xn0 - sx0[tt];
            const float dy = xn1 - sx1[tt];
            const float dz = xn2 - sx2[tt];
            const float pd = -(dx * dx + dy * dy + dz * dz);   // neg sq dist
            if (sm[tt] && pd > bestv[KNN_K - 1]) {
                int pos = 0;
#pragma unroll
                for (int j = 0; j < KNN_K; ++j) pos += (bestv[j] >= pd) ? 1 : 0;
#pragma unroll
                for (int j = KNN_K - 1; j > 0; --j) {
                    if (bestv[j - 1] < pd) { bestv[j] = bestv[j - 1]; besti[j] = besti[j - 1]; }
                }
                const int mm = m0 + tt;
#pragma unroll
                for (int j = 0; j < KNN_K; ++j)
                    if (j == pos) { bestv[j] = pd; besti[j] = mm; }
            }
        }
    }

    // gather neighbors, mean-center, 3x3 covariance
    float p0[KNN_K], p1[KNN_K], p2[KNN_K];
    float mu0 = 0.f, mu1 = 0.f, mu2 = 0.f;
#pragma unroll
    for (int j = 0; j < KNN_K; ++j) {
        const int idx = besti[j];
        p0[j] = xb[idx]; p1[j] = xb[PN_N + idx]; p2[j] = xb[2 * PN_N + idx];
        mu0 += p0[j]; mu1 += p1[j]; mu2 += p2[j];
    }
    const float inv = 1.0f / (float)KNN_K;
    mu0 *= inv; mu1 *= inv; mu2 *= inv;
    float c00 = 0.f, c01 = 0.f, c02 = 0.f, c11 = 0.f, c12 = 0.f, c22 = 0.f;
#pragma unroll
    for (int j = 0; j < KNN_K; ++j) {
        const float d0 = p0[j] - mu0, d1 = p1[j] - mu1, d2 = p2[j] - mu2;
        c00 += d0 * d0; c01 += d0 * d1; c02 += d0 * d2;
        c11 += d1 * d1; c12 += d1 * d2; c22 += d2 * d2;
    }
    const size_t M = PN_M;
    h0[0 * M + gid] = xn0;
    h0[1 * M + gid] = xn1;
    h0[2 * M + gid] = xn2;
    h0[3 * M + gid]  = c00; h0[4 * M + gid]  = c01; h0[5 * M + gid]  = c02;
    h0[6 * M + gid]  = c01; h0[7 * M + gid]  = c11; h0[8 * M + gid]  = c12;
    h0[9 * M + gid]  = c02; h0[10 * M + gid] = c12; h0[11 * M + gid] = c22;
}

// ---------- WMMA f32 GEMM: y[Cout,M] = W[Cout,Cin] @ act(h[Cin,M]) ----------
// act = relu(scale*h + shift) when BN_IN, else identity.
// Fused per-channel sum/sumsq atomics (BN stats of y); optional per-(b,c) min/max,
// optional skip of y store (layer 5).
template <bool BN_IN, bool STORE_Y, bool MINMAX>
__global__ void conv_gemm_kernel(const float* __restrict__ W, const float* __restrict__ h,
                                 const float* __restrict__ in_scale, const float* __restrict__ in_shift,
                                 float* __restrict__ y,
                                 float* __restrict__ osum, float* __restrict__ osumsq,
                                 unsigned* __restrict__ mmax, unsigned* __restrict__ mmin,
                                 int Cin, int Cout) {
    const int lane = threadIdx.x & 31;
    const int wave = threadIdx.x >> 5;
    const int wpb  = blockDim.x >> 5;
    const int coutBase = (blockIdx.y * wpb + wave) * 16;
    const int mBase = blockIdx.x * 16;
    const int half = lane >> 4;        // 0: K pair (0,1) / rows 0-7 of D; 1: K pair (2,3) / rows 8-15
    const int r = lane & 15;
    const int n = mBase + r;
    const size_t M = PN_M;

    v8f acc;
#pragma unroll
    for (int j = 0; j < 8; ++j) acc[j] = 0.0f;

    for (int k0 = 0; k0 < Cin; k0 += 4) {
        const int ka = k0 + 2 * half;
        v2f a;
        a[0] = W[(size_t)(coutBase + r) * Cin + ka];
        a[1] = W[(size_t)(coutBase + r) * Cin + ka + 1];
        float b0 = h[(size_t)ka * M + n];
        float b1 = h[(size_t)(ka + 1) * M + n];
        if (BN_IN) {
            b0 = fmaxf(0.0f, b0 * in_scale[ka]     + in_shift[ka]);
            b1 = fmaxf(0.0f, b1 * in_scale[ka + 1] + in_shift[ka + 1]);
        }
        v2f bb; bb[0] = b0; bb[1] = b1;
        acc = __builtin_amdgcn_wmma_f32_16x16x4_f32(false, a, false, bb, (short)0, acc,
                                                    false, false);
    }

    if (STORE_Y) {
#pragma unroll
        for (int j = 0; j < 8; ++j)
            y[(size_t)(coutBase + half * 8 + j) * M + n] = acc[j];
    }

    // per-channel sum / sumsq (reduce across the 16 lanes of each half)
    float s1[8], s2[8];
#pragma unroll
    for (int j = 0; j < 8; ++j) { s1[j] = acc[j]; s2[j] = acc[j] * acc[j]; }
#pragma unroll
    for (int msk = 1; msk <= 8; msk <<= 1) {
#pragma unroll
        for (int j = 0; j < 8; ++j) {
            s1[j] += __shfl_xor(s1[j], msk, 32);
            s2[j] += __shfl_xor(s2[j], msk, 32);
        }
    }
    if (r == 0) {
#pragma unroll
        for (int j = 0; j < 8; ++j) {
            const int ch = coutBase + half * 8 + j;
            atomicAdd(&osum[ch], s1[j]);
            atomicAdd(&osumsq[ch], s2[j]);
        }
    }

    if (MINMAX) {
        float mx[8], mn[8];
#pragma unroll
        for (int j = 0; j < 8; ++j) { mx[j] = acc[j]; mn[j] = acc[j]; }
#pragma unroll
        for (int msk = 1; msk <= 8; msk <<= 1) {
#pragma unroll
            for (int j = 0; j < 8; ++j) {
                mx[j] = fmaxf(mx[j], __shfl_xor(mx[j], msk, 32));
                mn[j] = fminf(mn[j], __shfl_xor(mn[j], msk, 32));
            }
        }
        if (r == 0) {
            const int bidx = mBase >> 13;   // / 8192 (tiles never straddle batches)
#pragma unroll
            for (int j = 0; j < 8; ++j) {
                const int ch = coutBase + half * 8 + j;
                atomicMax(&mmax[bidx * Cout + ch], enc_f32(mx[j]));
                atomicMin(&mmin[bidx * Cout + ch], enc_f32(mn[j]));
            }
        }
    }
}

// ---------- BN stat finalize: scale = g*rsqrt(var+eps), shift = b - mean*scale ----------
__global__ void finalize_stats_kernel(const float* __restrict__ osum, const float* __restrict__ osumsq,
                                      const float* __restrict__ g, const float* __restrict__ bb,
                                      float* __restrict__ scale, float* __restrict__ shift,
                                      int C, float invM) {
    const int c = blockIdx.x * blockDim.x + threadIdx.x;
    if (c >= C) return;
    const float m = osum[c] * invM;
    const float v = osumsq[c] * invM - m * m;
    const float s = g[c] * rsqrtf(v + BN_EPS);
    scale[c] = s;
    shift[c] = bb[c] - m * s;
}

// ---------- layer5 finalize + global max pool (BN affine is monotone per channel) ----------
__global__ void finalize_pool_kernel(const float* __restrict__ osum, const float* __restrict__ osumsq,
                                     const float* __restrict__ g, const float* __restrict__ bb,
                                     const unsigned* __restrict__ mmax, const unsigned* __restrict__ mmin,
                                     float* __restrict__ pooled, int C, float invM) {
    const int c = blockIdx.x * blockDim.x + threadIdx.x;
    if (c >= C) return;
    const float m = osum[c] * invM;
    const float v = osumsq[c] * invM - m * m;
    const float s = g[c] * rsqrtf(v + BN_EPS);
    const float t = bb[c] - m * s;
    for (int b = 0; b < PN_B; ++b) {
        const float ext = (s >= 0.0f) ? dec_f32(mmax[b * C + c]) : dec_f32(mmin[b * C + c]);
        pooled[b * C + c] = fmaxf(0.0f, s * ext + t);
    }
}

// ---------- FC1 (1024->512) + BN over batch + ReLU ----------
__global__ void fc1_kernel(const float* __restrict__ pooled, const float* __restrict__ lw1,
                           const float* __restrict__ g6, const float* __restrict__ b6,
                           float* __restrict__ hfc) {
    const int j = blockIdx.x * blockDim.x + threadIdx.x;
    const int EMB = 1024, OUT = 512;
    if (j >= OUT) return;
    const float* w = lw1 + (size_t)j * EMB;
    float a0 = 0.f, a1 = 0.f, a2 = 0.f, a3 = 0.f;
    for (int e = 0; e < EMB; ++e) {
        const float wv = w[e];
        a0 += pooled[e] * wv;
        a1 += pooled[EMB + e] * wv;
        a2 += pooled[2 * EMB + e] * wv;
        a3 += pooled[3 * EMB + e] * wv;
    }
    const float m = 0.25f * (a0 + a1 + a2 + a3);
    const float v = 0.25f * ((a0 - m) * (a0 - m) + (a1 - m) * (a1 - m) +
                             (a2 - m) * (a2 - m) + (a3 - m) * (a3 - m));
    const float s = g6[j] * rsqrtf(v + BN_EPS);
    const float t = b6[j] - m * s;
    hfc[0 * OUT + j] = fmaxf(0.0f, a0 * s + t);
    hfc[1 * OUT + j] = fmaxf(0.0f, a1 * s + t);
    hfc[2 * OUT + j] = fmaxf(0.0f, a2 * s + t);
    hfc[3 * OUT + j] = fmaxf(0.0f, a3 * s + t);
}

// ---------- FC2 (512->4) + bias ----------
__global__ void fc2_kernel(const float* __restrict__ hfc, const float* __restrict__ lw2,
                           const float* __restrict__ lb2, float* __restrict__ out) {
    const int t = threadIdx.x;
    if (t >= 16) return;
    const int b = t >> 2, o = t & 3;
    float acc = lb2[o];
    for (int e = 0; e < 512; ++e) acc += hfc[b * 512 + e] * lw2[o * 512 + e];
    out[b * 4 + o] = acc;
}

extern "C" void kernel_launch(void* const* d_in, const int* in_sizes, int n_in,
                              void* d_out, int out_size, void* d_ws, size_t ws_size,
                              hipStream_t stream) {
    const float* x          = (const float*)d_in[0];
    const unsigned char* li = (const unsigned char*)d_in[1];
    const float* w1 = (const float*)d_in[3];
    const float* w2 = (const float*)d_in[4];
    const float* w3 = (const float*)d_in[5];
    const float* w4 = (const float*)d_in[6];
    const float* w5 = (const float*)d_in[7];
    const float* g1 = (const float*)d_in[8],  *b1 = (const float*)d_in[9];
    const float* g2 = (const float*)d_in[10], *b2 = (const float*)d_in[11];
    const float* g3 = (const float*)d_in[12], *b3 = (const float*)d_in[13];
    const float* g4 = (const float*)d_in[14], *b4 = (const float*)d_in[15];
    const float* g5 = (const float*)d_in[16], *b5 = (const float*)d_in[17];
    const float* lw1 = (const float*)d_in[18];
    const float* g6 = (const float*)d_in[19], *b6 = (const float*)d_in[20];
    const float* lw2 = (const float*)d_in[21];
    const float* lb2 = (const float*)d_in[22];
    float* out = (float*)d_out;

    const size_t M = PN_M;
    float* wsf = (float*)d_ws;
    float* h0 = wsf;                     // 12*M
    float* y1 = h0 + 12 * M;             // 64*M
    float* y2 = y1 + 64 * M;             // 64*M
    float* y4 = y2 + 64 * M;             // 128*M
    float* y3 = y1;                      // alias (y1 dead after layer 2)
    float* stats = y4 + 128 * M;         // 2688 floats
    float* sum1 = stats + 0,    *sq1 = stats + 64;
    float* sum2 = stats + 128,  *sq2 = stats + 192;
    float* sum3 = stats + 256,  *sq3 = stats + 320;
    float* sum4 = stats + 384,  *sq4 = stats + 512;
    float* sum5 = stats + 640,  *sq5 = stats + 1664;
    float* ss = stats + 2688;            // 2688 floats (scale/shift)
    float* sc1 = ss + 0,    *sh1 = ss + 64;
    float* sc2 = ss + 128,  *sh2 = ss + 192;
    float* sc3 = ss + 256,  *sh3 = ss + 320;
    float* sc4 = ss + 384,  *sh4 = ss + 512;
    unsigned* mmax = (unsigned*)(ss + 2688);   // 4096
    unsigned* mmin = mmax + 4096;              // 4096
    float* pooled = (float*)(mmin + 4096);     // 4096
    float* hfc = pooled + 4096;                // 2048

    const float invM = 1.0f / (float)M;

    // zero BN accumulators + min/max sentinels (must happen every call)
    init_stats_kernel<<<dim3(16), dim3(256), 0, stream>>>(stats, 2688, mmax, mmin, 4096);

    // features: kNN + covariance -> h0 [12, M]
    knn_cov_kernel<<<dim3(PN_M / 256), dim3(256), 0, stream>>>(x, li, h0);

    // layer 1: 12 -> 64 (raw input)
    conv_gemm_kernel<false, true, false><<<dim3(M / 16, 1), dim3(128), 0, stream>>>(
        w1, h0, nullptr, nullptr, y1, sum1, sq1, nullptr, nullptr, 12, 64);
    finalize_stats_kernel<<<dim3(1), dim3(64), 0, stream>>>(sum1, sq1, g1, b1, sc1, sh1, 64, invM);

    // layer 2: 64 -> 64 (BN1+ReLU fused on load)
    conv_gemm_kernel<true, true, false><<<dim3(M / 16, 1), dim3(128), 0, stream>>>(
        w2, y1, sc1, sh1, y2, sum2, sq2, nullptr, nullptr, 64, 64);
    finalize_stats_kernel<<<dim3(1), dim3(64), 0, stream>>>(sum2, sq2, g2, b2, sc2, sh2, 64, invM);

    // layer 3: 64 -> 64
    conv_gemm_kernel<true, true, false><<<dim3(M / 16, 1), dim3(128), 0, stream>>>(
        w3, y2, sc2, sh2, y3, sum3, sq3, nullptr, nullptr, 64, 64);
    finalize_stats_kernel<<<dim3(1), dim3(64), 0, stream>>>(sum3, sq3, g3, b3, sc3, sh3, 64, invM);

    // layer 4: 64 -> 128
    conv_gemm_kernel<true, true, false><<<dim3(M / 16, 1), dim3(256), 0, stream>>>(
        w4, y3, sc3, sh3, y4, sum4, sq4, nullptr, nullptr, 64, 128);
    finalize_stats_kernel<<<dim3(1), dim3(128), 0, stream>>>(sum4, sq4, g4, b4, sc4, sh4, 128, invM);

    // layer 5: 128 -> 1024, fused stats + per-(b,c) min/max, y5 never materialized
    conv_gemm_kernel<true, false, true><<<dim3(M / 16, 8), dim3(256), 0, stream>>>(
        w5, y4, sc4, sh4, nullptr, sum5, sq5, mmax, mmin, 128, 1024);

    // BN5 + ReLU + global max pool via channel extrema
    finalize_pool_kernel<<<dim3(4), dim3(256), 0, stream>>>(sum5, sq5, g5, b5, mmax, mmin,
                                                           pooled, 1024, invM);

    // FC head
    fc1_kernel<<<dim3(2), dim3(256), 0, stream>>>(pooled, lw1, g6, b6, hfc);
    fc2_kernel<<<dim3(1), dim3(32), 0, stream>>>(hfc, lw2, lb2, out);

    (void)in_sizes; (void)n_in; (void)out_size; (void)ws_size;
}